// FeedForwardQuantum_65481071402351
// MI455X (gfx1250) — compile-verified
//
#include <hip/hip_runtime.h>

typedef __attribute__((ext_vector_type(2))) float v2f;
typedef __attribute__((ext_vector_type(8))) float v8f;

#define ROWS   (8 * 4096)   // B*S
#define EDIM   1024
#define NQ     8
#define DIM    256          // 2^NQ

// CNOT-ladder permutation: apply sigma_6 .. sigma_0 to index j.
// sigma_i flips target bit (qubit i+1, bitpos 6-i) if control bit (qubit i, bitpos 7-i) set.
__device__ __forceinline__ int qperm(int j) {
#pragma unroll
    for (int i = 6; i >= 0; --i)
        j ^= ((j >> (7 - i)) & 1) << (6 - i);
    return j;
}

// ---------------------------------------------------------------------------
// Kernel 1: fused  h = x.w1^T + b1  ->  8-qubit statevector sim  ->  q[row][8]
// One wave (32 lanes) per row; 8 waves / 256-thread block.
// LDS: w1 copy (32KB) + per-wave statevector (8 * 2KB).
// ---------------------------------------------------------------------------
__global__ __launch_bounds__(256) void ffq_quantum_kernel(
    const float* __restrict__ x, const float* __restrict__ w1,
    const float* __restrict__ b1, float* __restrict__ qout) {
    __shared__ __align__(16) float  w1s[NQ * EDIM];   // 32KB
    __shared__ float2 state[8][DIM];                  // 16KB

    const int lane = threadIdx.x & 31;
    const int wave = threadIdx.x >> 5;

    // stage w1 into LDS (float4 cooperative copy)
    {
        const float4* src = (const float4*)w1;
        float4*       dst = (float4*)w1s;
#pragma unroll
        for (int i = threadIdx.x; i < (NQ * EDIM) / 4; i += 256) dst[i] = src[i];
    }
    __syncthreads();

    const long  row = (long)blockIdx.x * 8 + wave;
    const float* xr = x + row * EDIM;

    // ---- h[f] = dot(x_row, w1[f]) ----
    float acc[NQ];
#pragma unroll
    for (int f = 0; f < NQ; ++f) acc[f] = 0.0f;
#pragma unroll
    for (int ch = 0; ch < EDIM / 128; ++ch) {
        const int k = ch * 128 + lane * 4;
        const float4 xv = *(const float4*)(xr + k);
#pragma unroll
        for (int f = 0; f < NQ; ++f) {
            const float* wp = &w1s[f * EDIM + k];
            acc[f] += xv.x * wp[0] + xv.y * wp[1] + xv.z * wp[2] + xv.w * wp[3];
        }
    }
#pragma unroll
    for (int f = 0; f < NQ; ++f)
#pragma unroll
        for (int off = 16; off > 0; off >>= 1)
            acc[f] += __shfl_xor(acc[f], off, 32);

    // ---- RX angles: single-qubit amps (cos(h/2), -i sin(h/2)) ----
    float cc[NQ], ss[NQ];
#pragma unroll
    for (int f = 0; f < NQ; ++f) {
        const float half = (acc[f] + b1[f]) * 0.5f;
        cc[f] = __cosf(half);
        ss[f] = __sinf(half);
    }

    // ---- product state: amp[j] = (prod c/s) * (-i)^popcount(j) ----
#pragma unroll
    for (int t = 0; t < 8; ++t) {
        const int j = t * 32 + lane;
        float m = 1.0f;
#pragma unroll
        for (int f = 0; f < NQ; ++f)
            m *= ((j >> (7 - f)) & 1) ? ss[f] : cc[f];
        const int p = __popc(j) & 3;
        const float re = (p == 0) ? m : ((p == 2) ? -m : 0.0f);
        const float im = (p == 3) ? m : ((p == 1) ? -m : 0.0f);
        state[wave][j] = make_float2(re, im);
    }
    __syncthreads();

    // ---- DEPTH=2 layers: gather via CNOT-ladder perm, then H on LSB qubit ----
    const float inv_sqrt2 = 0.70710678118654752f;
    float2 res[8];
#pragma unroll
    for (int L = 0; L < 2; ++L) {
#pragma unroll
        for (int t = 0; t < 8; ++t) {
            const int j  = t * 32 + lane;
            const int g0 = qperm(j & ~1);
            const int g1 = qperm(j | 1);
            const float2 v0 = state[wave][g0];
            const float2 v1 = state[wave][g1];
            const float sgn = (j & 1) ? -1.0f : 1.0f;
            res[t].x = (v0.x + sgn * v1.x) * inv_sqrt2;
            res[t].y = (v0.y + sgn * v1.y) * inv_sqrt2;
        }
        __syncthreads();
#pragma unroll
        for (int t = 0; t < 8; ++t) state[wave][t * 32 + lane] = res[t];
        __syncthreads();
    }

    // ---- <Z_f> = sum_j (1-2*bit_f(j)) * |amp_j|^2 ----
    float z[NQ];
#pragma unroll
    for (int f = 0; f < NQ; ++f) z[f] = 0.0f;
#pragma unroll
    for (int t = 0; t < 8; ++t) {
        const int j = t * 32 + lane;
        const float pr = res[t].x * res[t].x + res[t].y * res[t].y;
#pragma unroll
        for (int f = 0; f < NQ; ++f)
            z[f] += ((j >> (7 - f)) & 1) ? -pr : pr;
    }
#pragma unroll
    for (int f = 0; f < NQ; ++f)
#pragma unroll
        for (int off = 16; off > 0; off >>= 1)
            z[f] += __shfl_xor(z[f], off, 32);

    if (lane == 0) {
        float* qr = qout + row * NQ;
        *(float4*)(qr)     = make_float4(z[0], z[1], z[2], z[3]);
        *(float4*)(qr + 4) = make_float4(z[4], z[5], z[6], z[7]);
    }
}

// ---------------------------------------------------------------------------
// Kernel 2: out[r][e] = sum_f q[r][f]*w2[e][f] + b2[e]   (store-bound, 128MB)
// One 16x16 output tile per wave via two V_WMMA_F32_16X16X4_F32 (K=8).
// A layout (16x4 f32): lanes 0-15 -> M=lane, VGPR0=K0,VGPR1=K1; lanes 16-31 -> K2,K3.
// ---------------------------------------------------------------------------
__global__ __launch_bounds__(256) void ffq_out_wmma_kernel(
    const float* __restrict__ q, const float* __restrict__ w2,
    const float* __restrict__ b2, float* __restrict__ out) {
    const int lane = threadIdx.x & 31;
    const int wave = threadIdx.x >> 5;
    const int tile = blockIdx.x * 8 + wave;      // (ROWS/16)*(EDIM/16) tiles
    const int tn   = tile & (EDIM / 16 - 1);     // 64 col tiles
    const int tm   = tile / (EDIM / 16);         // 2048 row tiles

    const int mrow = lane & 15;                  // A row / B,D column within tile
    const int kb   = (lane < 16) ? 0 : 2;        // K pair held by this half-wave

    const float* qrow = q + (tm * 16 + mrow) * NQ;
    const v2f a_lo = { qrow[kb],     qrow[kb + 1] };
    const v2f a_hi = { qrow[kb + 4], qrow[kb + 5] };

    const int    n   = tn * 16 + mrow;
    const float* w2n = w2 + n * NQ;              // B[k][n] = w2[n*8 + k]
    const v2f b_lo = { w2n[kb],     w2n[kb + 1] };
    const v2f b_hi = { w2n[kb + 4], w2n[kb + 5] };

    v8f c = {};
    c = __builtin_amdgcn_wmma_f32_16x16x4_f32(false, a_lo, false, b_lo,
                                              (short)0, c, false, false);
    c = __builtin_amdgcn_wmma_f32_16x16x4_f32(false, a_hi, false, b_hi,
                                              (short)0, c, false, false);

    const float bias = b2[n];
    const int   m0   = tm * 16 + ((lane < 16) ? 0 : 8);
    float* orow = out + (long)m0 * EDIM + n;
#pragma unroll
    for (int v = 0; v < 8; ++v)                  // D VGPR v -> rows m0+v
        orow[(long)v * EDIM] = c[v] + bias;
}

extern "C" void kernel_launch(void* const* d_in, const int* in_sizes, int n_in,
                              void* d_out, int out_size, void* d_ws, size_t ws_size,
                              hipStream_t stream) {
    const float* x  = (const float*)d_in[0];
    const float* w1 = (const float*)d_in[1];
    const float* b1 = (const float*)d_in[2];
    const float* w2 = (const float*)d_in[3];
    const float* b2 = (const float*)d_in[4];
    float* out  = (float*)d_out;
    float* qbuf = (float*)d_ws;                  // ROWS*8 floats = 1 MB scratch

    ffq_quantum_kernel<<<ROWS / 8, 256, 0, stream>>>(x, w1, b1, qbuf);

    const int tiles = (ROWS / 16) * (EDIM / 16); // 131072
    ffq_out_wmma_kernel<<<tiles / 8, 256, 0, stream>>>(qbuf, w2, b2, out);
}